// seg_model_80350248174191
// MI455X (gfx1250) — compile-verified
//
#include <hip/hip_runtime.h>
#include <hip/hip_bf16.h>

// ---------------------------------------------------------------------------
// DGCNN part-segmentation forward for MI455X (gfx1250, wave32, WMMA + TDM).
// GEMMs on V_WMMA_F32_16X16X4_F32 (exact f32); feature tiles staged to LDS
// via TENSOR_LOAD_TO_LDS (Tensor Data Mover, TENSORcnt).
// ---------------------------------------------------------------------------

typedef __attribute__((ext_vector_type(8))) float v8f;
typedef __attribute__((ext_vector_type(2))) float v2f;
typedef __attribute__((ext_vector_type(4))) unsigned int v4u;
typedef __attribute__((ext_vector_type(8))) int v8i;
typedef __attribute__((ext_vector_type(4))) int v4i;

static constexpr int B_ = 4;
static constexpr int N_ = 4096;
static constexpr int KNN_ = 40;
static constexpr int NSEG_ = 6;

#define DEV __device__ __forceinline__

DEV v8f wmma4(v2f a, v2f b, v8f c) {
  // D = A(16x4 f32) * B(4x16 f32) + C(16x16 f32)
  return __builtin_amdgcn_wmma_f32_16x16x4_f32(
      /*neg_a=*/false, a, /*neg_b=*/false, b,
      /*c_mod=*/(short)0, c, /*reuse_a=*/false, /*reuse_b=*/false);
}

DEV float lrelu(float x) { return x > 0.f ? x : 0.2f * x; }

DEV v8f vzero8() { v8f z = {0.f,0.f,0.f,0.f,0.f,0.f,0.f,0.f}; return z; }

// ---------------------------------------------------------------------------
// TDM: DMA a 2D f32 tile (tileRows x rowLen, contiguous rows of a
// (tensorRows x rowLen) row-major tensor) from global into LDS.
// D# layout per CDNA5 ISA 8.3/8.4. Issue from ONE wave only; completion via
// s_wait_tensorcnt. TDM ignores EXEC, so a scalar branch gates issue.
// ---------------------------------------------------------------------------
DEV void tdm_load_2d_f32(const float* gsrc, unsigned int ldsByteAddr,
                         int tensorRows, int rowLen, int tileRows) {
  unsigned long long ga = (unsigned long long)(uintptr_t)gsrc;
  v4u g0;
  g0.x = 1u;                                                   // count=1, user D#
  g0.y = ldsByteAddr;                                          // lds_addr
  g0.z = (unsigned int)ga;                                     // global_addr lo
  g0.w = (unsigned int)((ga >> 32) & 0x01FFFFFFu) | (2u << 30);// addr hi | type=2
  v8i g1;
  g1[0] = (int)(2u << 16);                                     // data_size=4B
  g1[1] = (int)(((unsigned)rowLen & 0xFFFFu) << 16);           // tensor_dim0[15:0]
  g1[2] = (int)((((unsigned)tensorRows & 0xFFFFu) << 16) |     // tensor_dim1 lo
                (((unsigned)rowLen >> 16) & 0xFFFFu));         // | tensor_dim0 hi
  g1[3] = (int)((((unsigned)rowLen & 0xFFFFu) << 16) |         // tile_dim0
                (((unsigned)tensorRows >> 16) & 0xFFFFu));     // | tensor_dim1 hi
  g1[4] = (int)((unsigned)tileRows & 0xFFFFu);                 // tile_dim1 (dim2=0)
  g1[5] = rowLen;                                              // dim0_stride lo32
  g1[6] = 0;
  g1[7] = 0;
  v4i gz = {0, 0, 0, 0};
#if defined(__clang_major__) && (__clang_major__ >= 23)
  v8i gz8 = {0, 0, 0, 0, 0, 0, 0, 0};
  __builtin_amdgcn_tensor_load_to_lds(g0, g1, gz, gz, gz8, 0);
#else
  __builtin_amdgcn_tensor_load_to_lds(g0, g1, gz, gz, 0);
#endif
}

// ---------------------------------------------------------------------------
// sq[b][n] = sum_c X[b][n][c]^2     (X is (B,N,C) row-major)
// ---------------------------------------------------------------------------
template<int C>
__global__ void sqnorm_kernel(const float* __restrict__ X, float* __restrict__ sq) {
  int i = blockIdx.x * blockDim.x + threadIdx.x;
  if (i < B_ * N_) {
    const float* p = X + (size_t)i * C;
    float s = 0.f;
#pragma unroll
    for (int c = 0; c < C; ++c) s += p[c] * p[c];
    sq[i] = s;
  }
}

// ---------------------------------------------------------------------------
// KNN: workgroup owns 16 rows of one batch. Compute d = 2*X*X^T - sqi - sqj
// for 16 rows x all 4096 cols via WMMA into a 256KB LDS strip, then each
// wave extracts top-40 (ties -> lowest index) for 2 rows using cached
// per-lane argmax (only the winner's lane rescans its 128-element stride).
// Grid: B * (N/16) blocks, 256 threads (8 waves), dynLDS = 16*N*4 bytes.
// ---------------------------------------------------------------------------
template<int C>
__global__ __launch_bounds__(256) void knn_kernel(const float* __restrict__ X,
                                                  const float* __restrict__ sq,
                                                  int* __restrict__ idxOut) {
  extern __shared__ float strip[];           // [16][N_]
  __shared__ float sqRow[16];
  constexpr int CH = (C + 3) / 4;            // K chunks of 4

  const int b     = blockIdx.x / (N_ / 16);
  const int rbase = (blockIdx.x % (N_ / 16)) * 16;
  const int tid = threadIdx.x, lane = tid & 31, wave = tid >> 5;
  const int half = lane >> 4, l16 = lane & 15;

  const float* Xb  = X  + (size_t)b * N_ * C;
  const float* sqb = sq + (size_t)b * N_;
  if (tid < 16) sqRow[tid] = sqb[rbase + tid];

  // A tile: rows rbase..rbase+15. lanes0-15: K={kc,kc+1}; lanes16-31: K={kc+2,kc+3}
  v2f a[CH];
  {
    const float* ar = Xb + (size_t)(rbase + l16) * C + half * 2;
#pragma unroll
    for (int ch = 0; ch < CH; ++ch) {
      int ka = ch * 4 + half * 2;
      a[ch].x = (ka     < C) ? ar[ch * 4]     : 0.f;
      a[ch].y = (ka + 1 < C) ? ar[ch * 4 + 1] : 0.f;
    }
  }
  __syncthreads();

  // Distance strip via WMMA
  for (int t = wave; t < N_ / 16; t += 8) {
    const int cbase = t * 16;
    const float* br = Xb + (size_t)(cbase + l16) * C + half * 2;
    v8f acc = vzero8();
#pragma unroll
    for (int ch = 0; ch < CH; ++ch) {
      int ka = ch * 4 + half * 2;
      v2f bv;
      bv.x = (ka     < C) ? br[ch * 4]     : 0.f;
      bv.y = (ka + 1 < C) ? br[ch * 4 + 1] : 0.f;
      acc = wmma4(a[ch], bv, acc);
    }
    const float sqc = sqb[cbase + l16];
#pragma unroll
    for (int v = 0; v < 8; ++v) {
      int r = half * 8 + v;                  // D: VGPR v + half*8 = row, lane&15 = col
      strip[r * N_ + cbase + l16] = 2.f * acc[v] - sqRow[r] - sqc;
    }
  }
  __syncthreads();

  // Top-40 per row (2 rows per wave) with incremental per-lane argmax cache
  for (int rr = wave * 2; rr < wave * 2 + 2; ++rr) {
    float* drow = strip + rr * N_;
    int* orow = idxOut + ((size_t)b * N_ + rbase + rr) * KNN_;

    float lmax = -3.0e38f; int lidx = N_;
    for (int c = lane; c < N_; c += 32) {
      float v = drow[c];
      if (v > lmax) { lmax = v; lidx = c; }  // ascending scan keeps lowest idx on tie
    }
    for (int s = 0; s < KNN_; ++s) {
      float best = lmax; int bidx = lidx;
#pragma unroll
      for (int off = 16; off > 0; off >>= 1) {
        float ov = __shfl_xor(best, off, 32);
        int   oi = __shfl_xor(bidx, off, 32);
        if (ov > best || (ov == best && oi < bidx)) { best = ov; bidx = oi; }
      }
      if (lane == 0) orow[s] = bidx;
      if (lane == (bidx & 31)) {             // only owner lane invalidates + rescans
        drow[bidx] = -3.0e38f;
        lmax = -3.0e38f; lidx = N_;
        for (int c = lane; c < N_; c += 32) {
          float v = drow[c];
          if (v > lmax) { lmax = v; lidx = c; }
        }
      }
    }
  }
}

template<int CIN> DEV float edgef(const float* xj, const float* xi, int k) {
  if (k < CIN)      return xj[k] - xi[k];
  if (k < 2 * CIN)  return xi[k - CIN];
  return 0.f;
}

// ---------------------------------------------------------------------------
// Fused EdgeConv: gather -> [xj-xi, xi] -> W_a(2CIN->64)+LReLU
//                 [-> W_b(64->64)+LReLU] -> max over k -> out (B,N,64)
// Grid: B*(N/16) blocks of 128 threads (4 waves, one 16-Cout row tile each).
// ---------------------------------------------------------------------------
template<int CIN, bool TWO>
__global__ __launch_bounds__(128) void econv_kernel(const float* __restrict__ X,
                                                    const int* __restrict__ idx,
                                                    const float* __restrict__ Wa,
                                                    const float* __restrict__ ba,
                                                    const float* __restrict__ Wb,
                                                    const float* __restrict__ bb_,
                                                    float* __restrict__ out) {
  constexpr int K1  = 2 * CIN;
  constexpr int CH1 = (K1 + 3) / 4;
  __shared__ float h1[64 * 16];
  __shared__ float h2[64 * 16];
  __shared__ float omax[64 * 16];            // [cout][point]

  const int b     = blockIdx.x / (N_ / 16);
  const int pbase = (blockIdx.x % (N_ / 16)) * 16;
  const int tid = threadIdx.x, lane = tid & 31, wave = tid >> 5;
  const int half = lane >> 4, l16 = lane & 15;

  for (int e = tid; e < 1024; e += 128) omax[e] = -3.0e38f;

  // Preload A tiles (weights) and biases
  v2f a1[CH1];
  {
    const float* wr = Wa + (size_t)(wave * 16 + l16) * K1 + half * 2;
#pragma unroll
    for (int ch = 0; ch < CH1; ++ch) {
      int ka = ch * 4 + half * 2;
      a1[ch].x = (ka     < K1) ? wr[ch * 4]     : 0.f;
      a1[ch].y = (ka + 1 < K1) ? wr[ch * 4 + 1] : 0.f;
    }
  }
  v2f a2[TWO ? 16 : 1];
  float bias1[8], bias2[8];
#pragma unroll
  for (int v = 0; v < 8; ++v) bias1[v] = ba[wave * 16 + half * 8 + v];
  if constexpr (TWO) {
    const float* wr = Wb + (size_t)(wave * 16 + l16) * 64 + half * 2;
#pragma unroll
    for (int ch = 0; ch < 16; ++ch) { a2[ch].x = wr[ch * 4]; a2[ch].y = wr[ch * 4 + 1]; }
#pragma unroll
    for (int v = 0; v < 8; ++v) bias2[v] = bb_[wave * 16 + half * 8 + v];
  }
  __syncthreads();

  for (int nt = 0; nt < 40; ++nt) {
    const int gc = nt * 16 + l16;
    const int pl = gc / KNN_, kk = gc % KNN_;
    const int point = pbase + pl;
    const int j = idx[((size_t)b * N_ + point) * KNN_ + kk];
    const float* xj = X + ((size_t)b * N_ + j) * CIN;
    const float* xi = X + ((size_t)b * N_ + point) * CIN;

    v8f acc = vzero8();
#pragma unroll
    for (int ch = 0; ch < CH1; ++ch) {
      int k0 = ch * 4 + half * 2;
      v2f bv; bv.x = edgef<CIN>(xj, xi, k0); bv.y = edgef<CIN>(xj, xi, k0 + 1);
      acc = wmma4(a1[ch], bv, acc);
    }
    float* dst = TWO ? h1 : h2;
#pragma unroll
    for (int v = 0; v < 8; ++v) {
      int m = wave * 16 + half * 8 + v;
      dst[m * 16 + l16] = lrelu(acc[v] + bias1[v]);
    }
    __syncthreads();

    if constexpr (TWO) {
      v8f acc2 = vzero8();
#pragma unroll
      for (int ch = 0; ch < 16; ++ch) {
        int k0 = ch * 4 + half * 2;
        v2f bv; bv.x = h1[k0 * 16 + l16]; bv.y = h1[(k0 + 1) * 16 + l16];
        acc2 = wmma4(a2[ch], bv, acc2);
      }
#pragma unroll
      for (int v = 0; v < 8; ++v) {
        int m = wave * 16 + half * 8 + v;
        h2[m * 16 + l16] = lrelu(acc2[v] + bias2[v]);
      }
      __syncthreads();
    }

    if (tid < 64) {                          // thread t owns cout row t
      for (int col = 0; col < 16; ++col) {
        int pl2 = (nt * 16 + col) / KNN_;
        float v = h2[tid * 16 + col];
        float* om = &omax[tid * 16 + pl2];
        if (v > *om) *om = v;
      }
    }
    __syncthreads();
  }

  for (int e = tid; e < 1024; e += 128) {
    int pl = e >> 6, co = e & 63;
    out[((size_t)b * N_ + pbase + pl) * 64 + co] = omax[co * 16 + pl];
  }
}

// ---------------------------------------------------------------------------
// g[b][m] = lrelu( max_n (W6[m,:] . xc[b][:,n]) + b6[m] ), xc = [x1;x2;x3]
// Grid: B*16 blocks (64 rows each), 256 threads (8 waves; 4 row tiles x 2 col groups)
// ---------------------------------------------------------------------------
__global__ __launch_bounds__(256) void gmax_kernel(const float* __restrict__ x1,
                                                   const float* __restrict__ x2,
                                                   const float* __restrict__ x3,
                                                   const float* __restrict__ W6,
                                                   const float* __restrict__ b6,
                                                   float* __restrict__ g) {
  __shared__ float red[2][64];
  const int b = blockIdx.x >> 4;
  const int mBase = (blockIdx.x & 15) * 64;
  const int lane = threadIdx.x & 31, wave = threadIdx.x >> 5;
  const int half = lane >> 4, l16 = lane & 15;
  const int rt = wave & 3, cg = wave >> 2;

  v2f aw[48];
  {
    const float* wr = W6 + (size_t)(mBase + rt * 16 + l16) * 192 + half * 2;
#pragma unroll
    for (int ch = 0; ch < 48; ++ch) { aw[ch].x = wr[ch * 4]; aw[ch].y = wr[ch * 4 + 1]; }
  }
  v8f rmax;
#pragma unroll
  for (int v = 0; v < 8; ++v) rmax[v] = -3.0e38f;

  for (int t = cg; t < N_ / 16; t += 2) {
    const int n = t * 16 + l16;
    const float* p1 = x1 + ((size_t)b * N_ + n) * 64;
    const float* p2 = x2 + ((size_t)b * N_ + n) * 64;
    const float* p3 = x3 + ((size_t)b * N_ + n) * 64;
    v8f acc = vzero8();
#pragma unroll
    for (int ch = 0; ch < 48; ++ch) {
      int k0 = ch * 4 + half * 2;
      v2f bv;
      bv.x = (k0 < 64) ? p1[k0] : (k0 < 128) ? p2[k0 - 64] : p3[k0 - 128];
      int k1 = k0 + 1;
      bv.y = (k1 < 64) ? p1[k1] : (k1 < 128) ? p2[k1 - 64] : p3[k1 - 128];
      acc = wmma4(aw[ch], bv, acc);
    }
#pragma unroll
    for (int v = 0; v < 8; ++v) rmax[v] = fmaxf(rmax[v], acc[v]);
  }
#pragma unroll
  for (int off = 8; off > 0; off >>= 1)
#pragma unroll
    for (int v = 0; v < 8; ++v) rmax[v] = fmaxf(rmax[v], __shfl_xor(rmax[v], off, 32));
  if (l16 == 0) {
#pragma unroll
    for (int v = 0; v < 8; ++v) red[cg][rt * 16 + half * 8 + v] = rmax[v];
  }
  __syncthreads();
  if (threadIdx.x < 64) {
    int m = mBase + threadIdx.x;
    g[b * 1024 + m] = lrelu(fmaxf(red[0][threadIdx.x], red[1][threadIdx.x]) + b6[m]);
  }
}

// lf[b][o] = lrelu(label[b,:] . W7[o,:] + b7[o])
__global__ void label_kernel(const float* __restrict__ label, const float* __restrict__ W7,
                             const float* __restrict__ b7, float* __restrict__ lf) {
  int t = threadIdx.x;
  if (t < B_ * 64) {
    int b = t >> 6, o = t & 63;
    float s = b7[o];
#pragma unroll
    for (int c = 0; c < 16; ++c) s += label[b * 16 + c] * W7[o * 16 + c];
    lf[b * 64 + o] = lrelu(s);
  }
}

// ---------------------------------------------------------------------------
// Final fused MLP per 16-point tile: feat(1280) -> 256 -> 256 -> 128 -> 6
// x1/x2/x3 16x64 tiles DMA'd to LDS by the Tensor Data Mover (point-major);
// g/lf broadcast values staged by VALU. Grid: B*(N/16), 256 threads.
// dynLDS = (3*1024 + 1088 + 4096 + 4096 + 2048) floats = 57.6 KB
// ---------------------------------------------------------------------------
__global__ __launch_bounds__(256) void final_kernel(
    const float* __restrict__ x1, const float* __restrict__ x2, const float* __restrict__ x3,
    const float* __restrict__ g,  const float* __restrict__ lf,
    const float* __restrict__ Wp1, const float* __restrict__ bp1,
    const float* __restrict__ Wp2, const float* __restrict__ bp2,
    const float* __restrict__ Wp3, const float* __restrict__ bp3,
    const float* __restrict__ Wp4, float* __restrict__ out) {
  extern __shared__ float sm[];
  float* xt = sm;                      // [3][16][64]  (point-major tiles, TDM dest)
  float* bc = xt + 3 * 1024;           // [1088] = g(1024) ++ lf(64), batch-broadcast
  float* h1 = bc + 1088;               // [256][16]
  float* h2 = h1 + 256 * 16;           // [256][16]
  float* h3 = h2 + 256 * 16;           // [128][16]

  const int b     = blockIdx.x / (N_ / 16);
  const int pbase = (blockIdx.x % (N_ / 16)) * 16;
  const int tid = threadIdx.x, lane = tid & 31, wave = tid >> 5;
  const int half = lane >> 4, l16 = lane & 15;

  // TDM: DMA the three 16x64 f32 feature tiles into LDS (wave 0 issues only;
  // scalar branch gates issue since TDM ignores EXEC).
  if (wave == 0) {
    unsigned int ldsBase = (unsigned int)(uintptr_t)xt;   // addr[31:0] == LDS offset
    tdm_load_2d_f32(x1 + ((size_t)b * N_ + pbase) * 64, ldsBase,             N_, 64, 16);
    tdm_load_2d_f32(x2 + ((size_t)b * N_ + pbase) * 64, ldsBase + 1024 * 4,  N_, 64, 16);
    tdm_load_2d_f32(x3 + ((size_t)b * N_ + pbase) * 64, ldsBase + 2048 * 4,  N_, 64, 16);
    __builtin_amdgcn_s_wait_tensorcnt(0);
  }
  // Broadcast features (same for all 16 points of this batch)
  for (int e = tid; e < 1088; e += 256)
    bc[e] = (e < 1024) ? g[b * 1024 + e] : lf[b * 64 + (e - 1024)];
  __syncthreads();

  // feat(k, col) accessor: k<192 from point-major x tiles, else broadcast
  auto featKN = [&](int k, int nl) -> float {
    if (k < 192) return xt[(k >> 6) * 1024 + nl * 64 + (k & 63)];
    return bc[k - 192];
  };

  // GEMM1: 256x1280, two 16-row tiles per wave
  for (int rt = wave; rt < 16; rt += 8) {
    v8f acc = vzero8();
    const float* wr = Wp1 + (size_t)(rt * 16 + l16) * 1280 + half * 2;
#pragma unroll 4
    for (int ch = 0; ch < 320; ++ch) {
      v2f a; a.x = wr[ch * 4]; a.y = wr[ch * 4 + 1];
      int k0 = ch * 4 + half * 2;
      v2f bv; bv.x = featKN(k0, l16); bv.y = featKN(k0 + 1, l16);
      acc = wmma4(a, bv, acc);
    }
#pragma unroll
    for (int v = 0; v < 8; ++v) {
      int m = rt * 16 + half * 8 + v;
      h1[m * 16 + l16] = lrelu(acc[v] + bp1[m]);
    }
  }
  __syncthreads();

  // GEMM2: 256x256
  for (int rt = wave; rt < 16; rt += 8) {
    v8f acc = vzero8();
    const float* wr = Wp2 + (size_t)(rt * 16 + l16) * 256 + half * 2;
#pragma unroll 4
    for (int ch = 0; ch < 64; ++ch) {
      v2f a; a.x = wr[ch * 4]; a.y = wr[ch * 4 + 1];
      int k0 = ch * 4 + half * 2;
      v2f bv; bv.x = h1[k0 * 16 + l16]; bv.y = h1[(k0 + 1) * 16 + l16];
      acc = wmma4(a, bv, acc);
    }
#pragma unroll
    for (int v = 0; v < 8; ++v) {
      int m = rt * 16 + half * 8 + v;
      h2[m * 16 + l16] = lrelu(acc[v] + bp2[m]);
    }
  }
  __syncthreads();

  // GEMM3: 128x256, one 16-row tile per wave
  {
    int rt = wave;
    v8f acc = vzero8();
    const float* wr = Wp3 + (size_t)(rt * 16 + l16) * 256 + half * 2;
#pragma unroll 4
    for (int ch = 0; ch < 64; ++ch) {
      v2f a; a.x = wr[ch * 4]; a.y = wr[ch * 4 + 1];
      int k0 = ch * 4 + half * 2;
      v2f bv; bv.x = h2[k0 * 16 + l16]; bv.y = h2[(k0 + 1) * 16 + l16];
      acc = wmma4(a, bv, acc);
    }
#pragma unroll
    for (int v = 0; v < 8; ++v) {
      int m = rt * 16 + half * 8 + v;
      h3[m * 16 + l16] = lrelu(acc[v] + bp3[m]);
    }
  }
  __syncthreads();

  // GEMM4: 6x128 (tiny) on VALU; out is (B,N,6)
  if (tid < 96) {
    int col = tid / 6, o = tid % 6;
    float s = 0.f;
#pragma unroll 8
    for (int c = 0; c < 128; ++c) s += Wp4[o * 128 + c] * h3[c * 16 + col];
    out[((size_t)b * N_ + pbase + col) * NSEG_ + o] = s;
  }
}

// ---------------------------------------------------------------------------
extern "C" void kernel_launch(void* const* d_in, const int* in_sizes, int n_in,
                              void* d_out, int out_size, void* d_ws, size_t ws_size,
                              hipStream_t stream) {
  const float* points = (const float*)d_in[0];
  const float* label  = (const float*)d_in[1];
  const float* W1 = (const float*)d_in[2];  const float* b1 = (const float*)d_in[3];
  const float* W2 = (const float*)d_in[4];  const float* b2 = (const float*)d_in[5];
  const float* W3 = (const float*)d_in[6];  const float* b3 = (const float*)d_in[7];
  const float* W4 = (const float*)d_in[8];  const float* b4 = (const float*)d_in[9];
  const float* W5 = (const float*)d_in[10]; const float* b5 = (const float*)d_in[11];
  const float* W6 = (const float*)d_in[12]; const float* b6 = (const float*)d_in[13];
  const float* W7 = (const float*)d_in[14]; const float* b7 = (const float*)d_in[15];
  const float* Wp1 = (const float*)d_in[16]; const float* bp1 = (const float*)d_in[17];
  const float* Wp2 = (const float*)d_in[18]; const float* bp2 = (const float*)d_in[19];
  const float* Wp3 = (const float*)d_in[20]; const float* bp3 = (const float*)d_in[21];
  const float* Wp4 = (const float*)d_in[22];
  float* out = (float*)d_out;

  // Workspace layout (floats): sq | idx(int) | x1 | x2 | x3 | g | lf
  float* ws = (float*)d_ws;
  float* sq = ws;                                   // B*N
  int*   nn = (int*)(ws + B_ * N_);                 // B*N*KNN ints
  float* x1 = ws + B_ * N_ + B_ * N_ * KNN_;        // B*N*64
  float* x2 = x1 + (size_t)B_ * N_ * 64;
  float* x3 = x2 + (size_t)B_ * N_ * 64;
  float* gg = x3 + (size_t)B_ * N_ * 64;            // B*1024
  float* lf = gg + B_ * 1024;                       // B*64

  const int nbTiles = B_ * (N_ / 16);               // 1024
  const size_t knnLds = (size_t)16 * N_ * sizeof(float);                        // 256 KB
  const size_t finLds = (size_t)(3 * 1024 + 1088 + 4096 + 4096 + 2048) * sizeof(float);

  // Block 1: knn on raw points (C=3)
  sqnorm_kernel<3><<<(B_ * N_ + 255) / 256, 256, 0, stream>>>(points, sq);
  knn_kernel<3><<<nbTiles, 256, knnLds, stream>>>(points, sq, nn);
  econv_kernel<3, true><<<nbTiles, 128, 0, stream>>>(points, nn, W1, b1, W2, b2, x1);

  // Block 2: knn on x1 (C=64)
  sqnorm_kernel<64><<<(B_ * N_ + 255) / 256, 256, 0, stream>>>(x1, sq);
  knn_kernel<64><<<nbTiles, 256, knnLds, stream>>>(x1, sq, nn);
  econv_kernel<64, true><<<nbTiles, 128, 0, stream>>>(x1, nn, W3, b3, W4, b4, x2);

  // Block 3: knn on x2 (C=64), single conv
  sqnorm_kernel<64><<<(B_ * N_ + 255) / 256, 256, 0, stream>>>(x2, sq);
  knn_kernel<64><<<nbTiles, 256, knnLds, stream>>>(x2, sq, nn);
  econv_kernel<64, false><<<nbTiles, 128, 0, stream>>>(x2, nn, W5, b5, nullptr, nullptr, x3);

  // Global branch + label branch
  gmax_kernel<<<B_ * 16, 256, 0, stream>>>(x1, x2, x3, W6, b6, gg);
  label_kernel<<<1, 256, 0, stream>>>(label, W7, b7, lf);

  // Fused point MLP head
  final_kernel<<<nbTiles, 256, finLds, stream>>>(x1, x2, x3, gg, lf,
                                                 Wp1, bp1, Wp2, bp2, Wp3, bp3, Wp4, out);
}